// PoissonSolver_3049426780629
// MI455X (gfx1250) — compile-verified
//
#include <hip/hip_runtime.h>
#include <math.h>

// ---------------------------------------------------------------------------
// PINN Poisson loss on MI455X (gfx1250), fp32 WMMA path.
//   streams: V (value), D0, D1 (first dir-derivs), SS (sum of 2nd dir-derivs)
//   loss = mean((lap-f)^2) + 0.01*mean(u_bnd^2)
// ---------------------------------------------------------------------------

typedef float v2f __attribute__((ext_vector_type(2)));
typedef float v8f __attribute__((ext_vector_type(8)));

#define HID     128
#define STRIDE  132          // padded LDS row stride (floats) -> bank-conflict free
#define PTS     32           // points per block
#define WAVES   2            // wave32 x2
#define TPB     (WAVES * 32)
#define ALPHA_F 0.01f
#define PI_F    3.14159265358979323846f

// ---- WMMA f32 16x16x4 wrapper ---------------------------------------------
__device__ __forceinline__ v8f wmma_f32(v2f a, v2f b, v8f c) {
  // (neg_a, A, neg_b, B, c_mod, C, reuse_a, reuse_b)
  return __builtin_amdgcn_wmma_f32_16x16x4_f32(false, a, false, b, (short)0, c,
                                               false, false);
}

// A fragment: 16x4 f32.  lanes 0-15 -> row=lane, K=k0,k0+1 ; lanes 16-31 ->
// row=lane-16, K=k0+2,k0+3.  `buf` points at [32][STRIDE] stream slab in LDS.
__device__ __forceinline__ v2f load_afrag(const float* buf, int waveBase,
                                          int lane, int k0) {
  int pt = waveBase + (lane & 15);
  int k  = k0 + ((lane >> 4) << 1);
  const float* p = buf + pt * STRIDE + k;
  v2f a;
  a.x = p[0];
  a.y = p[1];
  return a;
}

// B fragment: 4x16 f32 tile of W (row-major [K=128][N=128]).
// lanes 0-15 -> K=k0,k0+1 ; lanes 16-31 -> K=k0+2,k0+3 ; col = jcol + lane%16.
__device__ __forceinline__ v2f load_bfrag(const float* __restrict__ W, int lane,
                                          int k0, int jcol) {
  int n = jcol + (lane & 15);
  int k = k0 + ((lane >> 4) << 1);
  v2f b;
  b.x = W[k * HID + n];
  b.y = W[(k + 1) * HID + n];
  return b;
}

#define V8F_ZERO {0.f,0.f,0.f,0.f,0.f,0.f,0.f,0.f}

// ---- hidden layer, 4 Taylor streams (interior kernel) ---------------------
__device__ __forceinline__ void layer4(const float* __restrict__ W,
                                       const float* __restrict__ bias,
                                       const float* inb, float* outb,
                                       int waveBase, int lane) {
  const float* inV  = inb + 0 * PTS * STRIDE;
  const float* inD0 = inb + 1 * PTS * STRIDE;
  const float* inD1 = inb + 2 * PTS * STRIDE;
  const float* inSS = inb + 3 * PTS * STRIDE;
  float* outV  = outb + 0 * PTS * STRIDE;
  float* outD0 = outb + 1 * PTS * STRIDE;
  float* outD1 = outb + 2 * PTS * STRIDE;
  float* outSS = outb + 3 * PTS * STRIDE;

  for (int j = 0; j < HID; j += 16) {
    v8f zv = V8F_ZERO, zd0 = V8F_ZERO, zd1 = V8F_ZERO, zss = V8F_ZERO;
#pragma unroll 4
    for (int k0 = 0; k0 < HID; k0 += 4) {
      v2f bf = load_bfrag(W, lane, k0, j);          // one B frag, 4 streams
      zv  = wmma_f32(load_afrag(inV,  waveBase, lane, k0), bf, zv);
      zd0 = wmma_f32(load_afrag(inD0, waveBase, lane, k0), bf, zd0);
      zd1 = wmma_f32(load_afrag(inD1, waveBase, lane, k0), bf, zd1);
      zss = wmma_f32(load_afrag(inSS, waveBase, lane, k0), bf, zss);
    }
    float bb  = bias[j + (lane & 15)];
    int   col = j + (lane & 15);
    int   rowOff = (lane >> 4) * 8;                 // C layout: hi lanes row+8
#pragma unroll
    for (int r = 0; r < 8; ++r) {
      float t   = tanhf(zv[r] + bb);
      float fp  = 1.f - t * t;                      // tanh'
      float fpp = -2.f * t * fp;                    // tanh''
      float a0 = zd0[r], a1 = zd1[r];
      int addr = (waveBase + rowOff + r) * STRIDE + col;
      outV[addr]  = t;
      outD0[addr] = fp * a0;
      outD1[addr] = fp * a1;
      outSS[addr] = fp * zss[r] + fpp * (a0 * a0 + a1 * a1);
    }
  }
}

// ---- hidden layer, value stream only (boundary kernel) --------------------
__device__ __forceinline__ void layer1(const float* __restrict__ W,
                                       const float* __restrict__ bias,
                                       const float* inb, float* outb,
                                       int waveBase, int lane) {
  for (int j = 0; j < HID; j += 16) {
    v8f zv = V8F_ZERO;
#pragma unroll 4
    for (int k0 = 0; k0 < HID; k0 += 4) {
      v2f bf = load_bfrag(W, lane, k0, j);
      zv = wmma_f32(load_afrag(inb, waveBase, lane, k0), bf, zv);
    }
    float bb  = bias[j + (lane & 15)];
    int   col = j + (lane & 15);
    int   rowOff = (lane >> 4) * 8;
#pragma unroll
    for (int r = 0; r < 8; ++r)
      outb[(waveBase + rowOff + r) * STRIDE + col] = tanhf(zv[r] + bb);
  }
}

// ---------------------------------------------------------------------------
__global__ __launch_bounds__(TPB) void pinn_interior(
    const float* __restrict__ x, int n,
    const float* __restrict__ W1, const float* __restrict__ b1,
    const float* __restrict__ W2, const float* __restrict__ b2,
    const float* __restrict__ W3, const float* __restrict__ b3,
    const float* __restrict__ W4, float* __restrict__ accum) {
  __shared__ float bufA[4 * PTS * STRIDE];
  __shared__ float bufB[4 * PTS * STRIDE];
  __shared__ float xs[PTS * 2];
  __shared__ float red[PTS];

  const int tid  = threadIdx.x;
  const int base = blockIdx.x * PTS;

  for (int i = tid; i < PTS * 2; i += TPB) {
    int g = base * 2 + i;
    xs[i] = x[g < n * 2 ? g : (n * 2 - 1)];
  }
  __syncthreads();

  // layer 1: 2 -> 128 (VALU, K too small for WMMA)
  for (int e = tid; e < PTS * HID; e += TPB) {
    int pt = e >> 7, nn = e & (HID - 1);
    float w0 = W1[nn], w1 = W1[HID + nn];
    float z  = xs[pt * 2] * w0 + xs[pt * 2 + 1] * w1 + b1[nn];
    float t  = tanhf(z);
    float fp = 1.f - t * t;
    float fpp = -2.f * t * fp;
    int addr = pt * STRIDE + nn;
    bufA[0 * PTS * STRIDE + addr] = t;
    bufA[1 * PTS * STRIDE + addr] = fp * w0;
    bufA[2 * PTS * STRIDE + addr] = fp * w1;
    bufA[3 * PTS * STRIDE + addr] = fpp * (w0 * w0 + w1 * w1);
  }
  __syncthreads();

  const int lane = tid & 31;
  const int waveBase = (tid >> 5) * 16;
  layer4(W2, b2, bufA, bufB, waveBase, lane);
  __syncthreads();
  layer4(W3, b3, bufB, bufA, waveBase, lane);
  __syncthreads();

  // head: lap = sum_k SS3[k]*W4[k]; squared error vs forcing
  if (tid < PTS) {
    const float* ss = bufA + 3 * PTS * STRIDE + tid * STRIDE;
    float lap = 0.f;
#pragma unroll 8
    for (int k = 0; k < HID; ++k) lap += ss[k] * W4[k];
    float x0 = xs[tid * 2], x1 = xs[tid * 2 + 1];
    float x1s = x1 * x1;
    float s0  = sinf(PI_F * x0);
    float f = -PI_F * PI_F * (1.f + 4.f * x1s) * s0 * sinf(PI_F * x1s) +
              2.f * PI_F * s0 * cosf(PI_F * x1s);
    float err = lap - f;
    red[tid] = (base + tid < n) ? err * err : 0.f;
  }
  __syncthreads();
  if (tid == 0) {
    float s = 0.f;
    for (int i = 0; i < PTS; ++i) s += red[i];
    atomicAdd(accum, s);
  }
}

__global__ __launch_bounds__(TPB) void pinn_boundary(
    const float* __restrict__ x, int n,
    const float* __restrict__ W1, const float* __restrict__ b1,
    const float* __restrict__ W2, const float* __restrict__ b2,
    const float* __restrict__ W3, const float* __restrict__ b3,
    const float* __restrict__ W4, const float* __restrict__ b4,
    float* __restrict__ accum) {
  __shared__ float bufA[PTS * STRIDE];
  __shared__ float bufB[PTS * STRIDE];
  __shared__ float xs[PTS * 2];
  __shared__ float red[PTS];

  const int tid  = threadIdx.x;
  const int base = blockIdx.x * PTS;

  for (int i = tid; i < PTS * 2; i += TPB) {
    int g = base * 2 + i;
    xs[i] = x[g < n * 2 ? g : (n * 2 - 1)];
  }
  __syncthreads();

  for (int e = tid; e < PTS * HID; e += TPB) {
    int pt = e >> 7, nn = e & (HID - 1);
    float z = xs[pt * 2] * W1[nn] + xs[pt * 2 + 1] * W1[HID + nn] + b1[nn];
    bufA[pt * STRIDE + nn] = tanhf(z);
  }
  __syncthreads();

  const int lane = tid & 31;
  const int waveBase = (tid >> 5) * 16;
  layer1(W2, b2, bufA, bufB, waveBase, lane);
  __syncthreads();
  layer1(W3, b3, bufB, bufA, waveBase, lane);
  __syncthreads();

  if (tid < PTS) {
    const float* h = bufA + tid * STRIDE;
    float u = b4[0];
#pragma unroll 8
    for (int k = 0; k < HID; ++k) u += h[k] * W4[k];
    red[tid] = (base + tid < n) ? u * u : 0.f;
  }
  __syncthreads();
  if (tid == 0) {
    float s = 0.f;
    for (int i = 0; i < PTS; ++i) s += red[i];
    atomicAdd(accum, s);
  }
}

__global__ void zero_accum(float* ws) {
  if (threadIdx.x < 2) ws[threadIdx.x] = 0.f;
}

__global__ void finalize(const float* __restrict__ ws, float* __restrict__ out,
                         int nInt, int nBnd) {
  if (threadIdx.x == 0)
    out[0] = ws[0] / (float)nInt + ALPHA_F * (ws[1] / (float)nBnd);
}

// ---------------------------------------------------------------------------
extern "C" void kernel_launch(void* const* d_in, const int* in_sizes, int n_in,
                              void* d_out, int out_size, void* d_ws,
                              size_t ws_size, hipStream_t stream) {
  const float* xi = (const float*)d_in[0];
  const float* xb = (const float*)d_in[1];
  const float* W1 = (const float*)d_in[2];
  const float* b1 = (const float*)d_in[3];
  const float* W2 = (const float*)d_in[4];
  const float* b2 = (const float*)d_in[5];
  const float* W3 = (const float*)d_in[6];
  const float* b3 = (const float*)d_in[7];
  const float* W4 = (const float*)d_in[8];
  const float* b4 = (const float*)d_in[9];
  float* ws  = (float*)d_ws;
  float* out = (float*)d_out;

  const int nInt = in_sizes[0] / 2;  // 131072
  const int nBnd = in_sizes[1] / 2;  // 16384
  const int gInt = (nInt + PTS - 1) / PTS;
  const int gBnd = (nBnd + PTS - 1) / PTS;

  zero_accum<<<1, 32, 0, stream>>>(ws);
  pinn_interior<<<gInt, TPB, 0, stream>>>(xi, nInt, W1, b1, W2, b2, W3, b3, W4,
                                          ws + 0);
  pinn_boundary<<<gBnd, TPB, 0, stream>>>(xb, nBnd, W1, b1, W2, b2, W3, b3, W4,
                                          b4, ws + 1);
  finalize<<<1, 32, 0, stream>>>(ws, out, nInt, nBnd);
}